// CustomGCN_68702296867065
// MI455X (gfx1250) — compile-verified
//
#include <hip/hip_runtime.h>
#include <math.h>

// ---------------------------------------------------------------------------
// CDNA5 / gfx1250 GCN pipeline.
//
// Shapes: N=100000 nodes, E=1.6M edges, D=64 hidden, G=64 graphs, D_OUT=32.
//
// Exploits: emb vocab == 1  =>  h0 is the SAME vector v0 for every node,
// so layer 1 is rank-1:  h1[i] = relu(v * s_i + b1),
//   v   = emb[0] @ W1                                (64-vector)
//   s_i = dinv_i * (dinv_i + sum_{src->i} dinv_src)  (scalar per node)
// Layer 2 needs full 64-dim aggregation; we do it atomic-free via CSR.
// ---------------------------------------------------------------------------

typedef __attribute__((ext_vector_type(2))) float v2f;
typedef __attribute__((ext_vector_type(4))) float v4f;
typedef __attribute__((ext_vector_type(8))) float v8f;

__device__ inline v8f wmma_f32_16x16x4(v2f a, v2f b, v8f c) {
#if defined(__gfx1250__) && __has_builtin(__builtin_amdgcn_wmma_f32_16x16x4_f32)
  // (neg_a, A, neg_b, B, c_mod, C, reuse_a, reuse_b)
  return __builtin_amdgcn_wmma_f32_16x16x4_f32(false, a, false, b, (short)0, c,
                                               false, false);
#else
  return c; // host-side / fallback stub (never used on gfx1250)
#endif
}

// ---------------------------------------------------------------- utilities
__global__ void zero_u32_kernel(unsigned int* __restrict__ p, long n) {
  long i = (long)blockIdx.x * 256 + threadIdx.x;
  if (i < n) p[i] = 0u;
}

// deg histogram on dst (float counts; exact for deg < 2^24)
__global__ void edge_deg_kernel(const int* __restrict__ dst,
                                float* __restrict__ cntf, int E) {
  int e = blockIdx.x * 256 + threadIdx.x;
  if (e < E) atomicAdd(&cntf[dst[e]], 1.0f);
}

__global__ void dinv_kernel(const float* __restrict__ cntf,
                            float* __restrict__ dinv, int n) {
  int i = blockIdx.x * 256 + threadIdx.x;
  if (i < n) dinv[i] = rsqrtf(cntf[i] + 1.0f); // +1 self loop
}

// t[i] = sum over in-edges of dinv[src]
__global__ void edge_t_kernel(const int* __restrict__ src,
                              const int* __restrict__ dst,
                              const float* __restrict__ dinv,
                              float* __restrict__ t, int E) {
  int e = blockIdx.x * 256 + threadIdx.x;
  if (e < E) atomicAdd(&t[dst[e]], dinv[src[e]]);
}

// single-block exclusive scan of float counts -> rowptr[N+1]
__global__ void scan_rowptr_kernel(const float* __restrict__ cntf,
                                   int* __restrict__ rowptr, int n) {
  __shared__ int sh[1024];
  int carry = 0;
  if (threadIdx.x == 0) rowptr[0] = 0;
  for (int base = 0; base < n; base += 1024) {
    int i = base + (int)threadIdx.x;
    int val = (i < n) ? (int)cntf[i] : 0;
    sh[threadIdx.x] = val;
    __syncthreads();
    for (int off = 1; off < 1024; off <<= 1) {
      int tv = (threadIdx.x >= (unsigned)off) ? sh[threadIdx.x - off] : 0;
      __syncthreads();
      sh[threadIdx.x] += tv;
      __syncthreads();
    }
    if (i < n) rowptr[i + 1] = carry + sh[threadIdx.x];
    carry += sh[1023];
    __syncthreads();
  }
}

__global__ void csr_fill_kernel(const int* __restrict__ src,
                                const int* __restrict__ dst,
                                const int* __restrict__ rowptr,
                                int* __restrict__ fill,
                                int* __restrict__ colidx, int E) {
  int e = blockIdx.x * 256 + threadIdx.x;
  if (e < E) {
    int d = dst[e];
    int r = atomicAdd(&fill[d], 1);
    colidx[rowptr[d] + r] = src[e];
  }
}

// v = emb[0] @ W1   (tiny: 64 threads)
__global__ void compute_v_kernel(const float* __restrict__ emb,
                                 const float* __restrict__ W1,
                                 float* __restrict__ v) {
  int d = threadIdx.x;
  float acc = 0.f;
  for (int k = 0; k < 64; ++k) acc += emb[k] * W1[k * 64 + d];
  v[d] = acc;
}

// ---------------------------------------------------------------------------
// u[i][:] = (relu(v * s_i + b1) @ W2) * dinv_i
// One wave per 16-node tile; f32 WMMA 16x16x4, K=64 in 16 steps, 4 col-tiles.
// A layout: lane<16 holds M=lane, K pair (k0,k0+1); lane>=16 K pair (k0+2,k0+3)
// B layout: lane<16 holds N=lane, rows (k0,k0+1); lane>=16 rows (k0+2,k0+3)
// D layout: VGPR j -> row j (lanes 0-15) / row j+8 (lanes 16-31), col = lane%16
// ---------------------------------------------------------------------------
__global__ void node_u_kernel(const float* __restrict__ dinv,
                              const float* __restrict__ t,
                              const float* __restrict__ vvec,
                              const float* __restrict__ b1,
                              const float* __restrict__ W2,
                              float* __restrict__ u, int n) {
  __shared__ float w2s[64 * 65]; // pad stride 65 -> no bank conflicts
  __shared__ float vs[64];
  __shared__ float b1s[64];
  for (int i = threadIdx.x; i < 64 * 64; i += 256)
    w2s[(i >> 6) * 65 + (i & 63)] = W2[i];
  if (threadIdx.x < 64) {
    vs[threadIdx.x] = vvec[threadIdx.x];
    b1s[threadIdx.x] = b1[threadIdx.x];
  }
  __syncthreads();

  const int wave = threadIdx.x >> 5;
  const int lane = threadIdx.x & 31;
  const int h = lane >> 4;   // lane half selects K pair / output row half
  const int m = lane & 15;   // row (A) / col (B,D) within tile
  const int tileBase = (blockIdx.x * 8 + wave) * 16;
  if (tileBase >= n) return;

  int nodeM = tileBase + m;
  int nodeMc = nodeM < n ? nodeM : (n - 1);
  float di = dinv[nodeMc];
  float s = di * (di + t[nodeMc]); // rank-1 layer-1 scalar

  float dvr[8];
#pragma unroll
  for (int j = 0; j < 8; ++j) {
    int r = tileBase + j + 8 * h;
    dvr[j] = dinv[r < n ? r : (n - 1)];
  }

#pragma unroll
  for (int nt = 0; nt < 4; ++nt) {
    v8f c = {};
#pragma unroll
    for (int kb = 0; kb < 16; ++kb) {
      int k = kb * 4 + 2 * h;
      v2f a, b;
      float p0 = vs[k] * s + b1s[k];
      float p1 = vs[k + 1] * s + b1s[k + 1];
      a.x = p0 > 0.f ? p0 : 0.f; // h1 = relu(v*s + b1), built on the fly
      a.y = p1 > 0.f ? p1 : 0.f;
      b.x = w2s[k * 65 + nt * 16 + m];
      b.y = w2s[(k + 1) * 65 + nt * 16 + m];
      c = wmma_f32_16x16x4(a, b, c);
    }
#pragma unroll
    for (int j = 0; j < 8; ++j) {
      int r = tileBase + j + 8 * h;
      if (r < n) u[(long)r * 64 + nt * 16 + m] = c[j] * dvr[j];
    }
  }
}

// ---------------------------------------------------------------------------
// CSR aggregation + self term + relu + fused mean-pool accumulation.
// 16 lanes per node, each lane owns a float4 slab of the 64-dim feature.
// h2[i] = relu(dinv_i * (sum_{src->i} u[src] + u[i]) + b2)
// ---------------------------------------------------------------------------
__global__ void agg_pool_kernel(const int* __restrict__ rowptr,
                                const int* __restrict__ colidx,
                                const float* __restrict__ u,
                                const float* __restrict__ dinv,
                                const float* __restrict__ b2,
                                const int* __restrict__ batch,
                                float* __restrict__ pooled,
                                int* __restrict__ cntg, int n) {
  long tid = (long)blockIdx.x * 256 + threadIdx.x;
  int node = (int)(tid >> 4);
  int q = (int)(tid & 15);
  if (node >= n) return;
  int start = rowptr[node], end = rowptr[node + 1];
  v4f acc = {};
  for (int e = start; e < end; ++e) {
    int col = colidx[e]; // broadcast across the 16 lanes of this node
    v4f x = *(const v4f*)(u + (long)col * 64 + q * 4);
    acc += x;
  }
  v4f uself = *(const v4f*)(u + (long)node * 64 + q * 4);
  float di = dinv[node];
  int g = batch[node];
#pragma unroll
  for (int r = 0; r < 4; ++r) {
    float pre = di * (acc[r] + uself[r]) + b2[q * 4 + r];
    float hh = pre > 0.f ? pre : 0.f;
    atomicAdd(&pooled[g * 64 + q * 4 + r], hh);
  }
  if (q == 0) atomicAdd(&cntg[g], 1);
}

// ---------------------------------------------------------------------------
// out[G x 32] = (pooled / cnt) @ fcW + fcb   — single block, 8 WMMA tiles.
// ---------------------------------------------------------------------------
__global__ void final_gemm_kernel(const float* __restrict__ pooled,
                                  const int* __restrict__ cntg,
                                  const float* __restrict__ fcW,
                                  const float* __restrict__ fcb,
                                  float* __restrict__ out, int G) {
  __shared__ float ps[64 * 65];
  for (int i = threadIdx.x; i < 64 * 64; i += 256) {
    int g = i >> 6, d = i & 63;
    float val = 0.f;
    if (g < G) {
      int cn = cntg[g];
      val = pooled[i] / (float)(cn > 0 ? cn : 1);
    }
    ps[g * 65 + d] = val;
  }
  __syncthreads();

  int wave = threadIdx.x >> 5, lane = threadIdx.x & 31;
  int h = lane >> 4, m = lane & 15;
  int mt = wave >> 1, nt = wave & 1; // 4 row-tiles x 2 col-tiles
  int col = nt * 16 + m;
  v8f c = {};
#pragma unroll
  for (int kb = 0; kb < 16; ++kb) {
    int k = kb * 4 + 2 * h;
    v2f a, b;
    a.x = ps[(mt * 16 + m) * 65 + k];
    a.y = ps[(mt * 16 + m) * 65 + k + 1];
    b.x = fcW[k * 32 + col];
    b.y = fcW[(k + 1) * 32 + col];
    c = wmma_f32_16x16x4(a, b, c);
  }
  float bias = fcb[col];
#pragma unroll
  for (int j = 0; j < 8; ++j) {
    int row = mt * 16 + j + 8 * h;
    if (row < G) out[row * 32 + col] = c[j] + bias;
  }
}

// ---------------------------------------------------------------------------
extern "C" void kernel_launch(void* const* d_in, const int* in_sizes, int n_in,
                              void* d_out, int out_size, void* d_ws,
                              size_t ws_size, hipStream_t stream) {
  (void)n_in; (void)ws_size;
  const int N = in_sizes[0];
  const int E = in_sizes[1] / 2;
  const int G = out_size / 32;

  const int* edge = (const int*)d_in[1];
  const int* src = edge;
  const int* dst = edge + E;
  const int* batch = (const int*)d_in[2];
  const float* emb = (const float*)d_in[3];
  const float* W1 = (const float*)d_in[4];
  const float* b1 = (const float*)d_in[5];
  const float* W2 = (const float*)d_in[6];
  const float* b2 = (const float*)d_in[7];
  const float* fcW = (const float*)d_in[8];
  const float* fcb = (const float*)d_in[9];
  float* out = (float*)d_out;

  // workspace carve (256B aligned slabs)
  char* base = (char*)d_ws;
  size_t off = 0;
  auto carve = [&](size_t bytes) -> void* {
    void* p = base + off;
    off = (off + bytes + 255) & ~(size_t)255;
    return p;
  };
  float* cntf   = (float*)carve((size_t)N * 4);
  float* tbuf   = (float*)carve((size_t)N * 4);
  float* dinv   = (float*)carve((size_t)N * 4);
  float* vvec   = (float*)carve(64 * 4);
  int*   rowptr = (int*)carve((size_t)(N + 1) * 4);
  int*   colidx = (int*)carve((size_t)E * 4);
  int*   fill   = (int*)carve((size_t)N * 4);
  float* pooled = (float*)carve((size_t)G * 64 * 4);
  int*   cntg   = (int*)carve((size_t)G * 4);
  float* u      = (float*)carve((size_t)N * 64 * 4); // 25.6 MB, L2-resident

  auto gblk = [](long n) { return (unsigned)((n + 255) / 256); };

  // zero accumulators (harness does not re-initialize between replays)
  zero_u32_kernel<<<gblk(N), 256, 0, stream>>>((unsigned*)cntf, N);
  zero_u32_kernel<<<gblk(N), 256, 0, stream>>>((unsigned*)tbuf, N);
  zero_u32_kernel<<<gblk(N), 256, 0, stream>>>((unsigned*)fill, N);
  zero_u32_kernel<<<gblk((long)G * 64), 256, 0, stream>>>((unsigned*)pooled,
                                                          (long)G * 64);
  zero_u32_kernel<<<gblk(G), 256, 0, stream>>>((unsigned*)cntg, G);

  // degrees -> dinv -> t (rank-1 layer-1 aggregation, scalars only)
  edge_deg_kernel<<<gblk(E), 256, 0, stream>>>(dst, cntf, E);
  dinv_kernel<<<gblk(N), 256, 0, stream>>>(cntf, dinv, N);
  edge_t_kernel<<<gblk(E), 256, 0, stream>>>(src, dst, dinv, tbuf, E);

  // CSR build
  scan_rowptr_kernel<<<1, 1024, 0, stream>>>(cntf, rowptr, N);
  csr_fill_kernel<<<gblk(E), 256, 0, stream>>>(src, dst, rowptr, fill, colidx,
                                               E);

  // v = emb0 @ W1
  compute_v_kernel<<<1, 64, 0, stream>>>(emb, W1, vvec);

  // u = (relu(v*s + b1) @ W2) * dinv   — WMMA kernel
  int tiles = (N + 15) / 16;
  node_u_kernel<<<(unsigned)((tiles + 7) / 8), 256, 0, stream>>>(
      dinv, tbuf, vvec, b1, W2, u, N);

  // layer-2 aggregation + relu + fused mean-pool accumulation
  agg_pool_kernel<<<gblk((long)N * 16), 256, 0, stream>>>(
      rowptr, colidx, u, dinv, b2, batch, pooled, cntg, N);

  // out = (pooled / cnt) @ fcW + fcb — WMMA kernel
  final_gemm_kernel<<<1, 256, 0, stream>>>(pooled, cntg, fcW, fcb, out, G);
}